// LengthRegulator_59571196396010
// MI455X (gfx1250) — compile-verified
//
#include <hip/hip_runtime.h>
#include <stdint.h>

#define B 32
#define S 1024
#define D 512
#define D4 (D / 4)   // 128 float4 per frame

typedef __attribute__((ext_vector_type(4))) float f4;

// ---------------------------------------------------------------------------
// Kernel 0: detect durations dtype layout (int32 vs int64-little-endian).
// For int64 values in [0,8), every high 32-bit word is 0. For int32 data,
// the probability that all 16384 sampled odd words are 0 is 8^-16384 ~ 0.
// Only touches the first B*S 32-bit words -> in-bounds for both layouts.
// flag[0] = 1 -> int32 layout, 0 -> int64 layout. Deterministic per input.
// ---------------------------------------------------------------------------
__global__ __launch_bounds__(256) void lr_detect_kernel(
    const unsigned* __restrict__ raw, int* __restrict__ flag) {
  __shared__ int any;
  if (threadIdx.x == 0) any = 0;
  __syncthreads();
  int local = 0;
  for (int i = threadIdx.x; i < (B * S) / 2; i += 256)
    local |= (raw[2 * i + 1] != 0u) ? 1 : 0;
  if (local) atomicOr(&any, 1);
  __syncthreads();
  if (threadIdx.x == 0) flag[0] = any;
}

// ---------------------------------------------------------------------------
// Kernel 1: per-row inclusive scan of durations.
// One 1024-thread block per row. The row is staged into LDS with the gfx1250
// async global->LDS copy (tracked by ASYNCcnt), then scanned in LDS.
// ---------------------------------------------------------------------------
__global__ __launch_bounds__(1024) void lr_scan_kernel(
    const unsigned* __restrict__ raw, const int* __restrict__ flag,
    int* __restrict__ cums, int* __restrict__ totals) {
  __shared__ int s[S];
  const int b = blockIdx.x;
  const int tid = threadIdx.x;
  const int g = b * S + tid;
  // int32 layout: element g at raw[g]; int64 layout: low word at raw[2*g].
  const unsigned* src = (flag[0] != 0) ? (raw + g) : (raw + 2 * g);
#if defined(__gfx1250__)
  {
    // Per-lane async copy of one dword into this thread's LDS slot.
    // LDS address comes from the vdst VGPR; flat shared ptr low 32 bits == LDS offset.
    uint32_t lds_off = (uint32_t)(uintptr_t)(&s[tid]);
    uint64_t gaddr   = (uint64_t)(uintptr_t)(src);
    asm volatile("global_load_async_to_lds_b32 %0, %1, off"
                 :: "v"(lds_off), "v"(gaddr) : "memory");
    asm volatile("s_wait_asynccnt 0" ::: "memory");
  }
#else
  s[tid] = (int)(*src);
#endif
  __syncthreads();
  // Hillis-Steele inclusive scan over 1024 elements.
  for (int off = 1; off < S; off <<= 1) {
    int v = 0;
    if (tid >= off) v = s[tid - off];
    __syncthreads();
    s[tid] += v;
    __syncthreads();
  }
  cums[g] = s[tid];
  if (tid == S - 1) totals[b] = s[tid];
}

// ---------------------------------------------------------------------------
// Kernel 2: idx[b,t] = searchsorted(cums[b], t, side='right'), clipped;
// -1 sentinel where t >= totals[b] (zero-pad region).
// cums (128 KB) is L2-resident -> the 10-step binary search is cheap.
// ---------------------------------------------------------------------------
__global__ __launch_bounds__(256) void lr_index_kernel(
    const int* __restrict__ cums, const int* __restrict__ totals,
    int* __restrict__ idx, int T) {
  const int t = blockIdx.x * 256 + threadIdx.x;
  const int b = blockIdx.y;
  if (t >= T) return;
  int out;
  if (t >= totals[b]) {
    out = -1;
  } else {
    const int* c = cums + b * S;
    int lo = 0, hi = S;
    while (lo < hi) {
      int mid = (lo + hi) >> 1;
      if (c[mid] <= t) lo = mid + 1; else hi = mid;
    }
    out = (lo < S) ? lo : (S - 1);
  }
  idx[b * T + t] = out;
}

// ---------------------------------------------------------------------------
// Kernel 3: streaming gather. One float4 (16B) per thread, fully coalesced.
// x reads stay cached (x = 64 MB fits in the 192 MB L2; each frame is reused
// ~3.5x). Output (~236 MB) is written with non-temporal B128 stores so the
// streaming writes don't evict x from L2.
// ---------------------------------------------------------------------------
__global__ __launch_bounds__(256) void lr_gather_kernel(
    const f4* __restrict__ x4, const int* __restrict__ idx,
    f4* __restrict__ out4, int T) {
  const int b = blockIdx.y;
  const int flat = blockIdx.x * 256 + threadIdx.x;   // [0, T*D4)
  if (flat >= T * D4) return;
  const int t = flat >> 7;          // D4 == 128
  const int q = flat & (D4 - 1);
  const int src = idx[b * T + t];   // broadcast across the 128 threads of a frame
  f4 v = {0.0f, 0.0f, 0.0f, 0.0f};
  if (src >= 0) v = x4[(b * S + src) * D4 + q];
  __builtin_nontemporal_store(v, &out4[(b * T + t) * D4 + q]);
}

// ---------------------------------------------------------------------------
extern "C" void kernel_launch(void* const* d_in, const int* in_sizes, int n_in,
                              void* d_out, int out_size, void* d_ws, size_t ws_size,
                              hipStream_t stream) {
  (void)in_sizes; (void)n_in; (void)ws_size;
  const f4*       x4  = (const f4*)d_in[0];
  const unsigned* dur = (const unsigned*)d_in[1];
  f4*             out = (f4*)d_out;

  const int T = out_size / (B * D);   // out_size = B*T*D (float32 elements)

  // Workspace layout (ints): [flag(16 pad)] [cums B*S] [totals B] [idx B*T]
  int* wsI    = (int*)d_ws;
  int* flag   = wsI;
  int* cums   = wsI + 16;
  int* totals = cums + B * S;
  int* idx    = totals + B;

  lr_detect_kernel<<<1, 256, 0, stream>>>(dur, flag);
  lr_scan_kernel<<<B, S, 0, stream>>>(dur, flag, cums, totals);

  dim3 gIdx((T + 255) / 256, B);
  lr_index_kernel<<<gIdx, 256, 0, stream>>>(cums, totals, idx, T);

  dim3 gCopy((T * D4 + 255) / 256, B);
  lr_gather_kernel<<<gCopy, 256, 0, stream>>>(x4, idx, out, T);
}